// ReidBench_2783138808143
// MI455X (gfx1250) — compile-verified
//
#include <hip/hip_runtime.h>
#include <hip/hip_bf16.h>
#include <stdint.h>

// ---------------- problem constants ----------------
#define NCLS   90
#define NA     9
#define KDET   5000
#define NBATCH 2
#define CCLS   810              // NA*NCLS
#define PTOT   5456             // sum of level spatial sizes
#define MPB    (CCLS * PTOT)    // 4,419,360 scores per batch
#define CAP    16384            // candidate buffer cap
#define CLS_CH 7290             // 3 * NA*NCLS*NGMM
#define BOX_CH 324              // 3 * NA*4*NGMM
#define BK     (NBATCH * KDET)  // 10000

struct LvlPtrs {
    const float* cls[5];
    const float* box[5];
};

// order-preserving float -> uint key (descending float == descending key)
__device__ __forceinline__ uint32_t f2key(float f) {
    uint32_t u = __float_as_uint(f);
    return ((int32_t)u < 0) ? ~u : (u | 0x80000000u);
}

// ---- CDNA5 async global -> LDS copies (per-lane, tracked by ASYNCcnt) ----
__device__ __forceinline__ void async_g2l_b128(uint32_t lds_off, const float* gptr) {
    asm volatile("global_load_async_to_lds_b128 %0, %1, off"
                 :: "v"(lds_off), "v"((uint64_t)(uintptr_t)gptr) : "memory");
}
__device__ __forceinline__ void wait_async_zero() {
    asm volatile("s_wait_asynccnt 0" ::: "memory");
}

__device__ __forceinline__ float sigm(float x) { return 1.0f / (1.0f + expf(-x)); }

__device__ __forceinline__ uint32_t gmm_key(float m0, float m1, float m2,
                                            float w0, float w1, float w2) {
    float wmax = fmaxf(w0, fmaxf(w1, w2));
    float e0 = expf(w0 - wmax), e1 = expf(w1 - wmax), e2 = expf(w2 - wmax);
    float inv = 1.0f / (e0 + e1 + e2);
    float wm = (e0 * m0 + e1 * m1 + e2 * m2) * inv;
    return f2key(wm);
}

// ---------------- K1: class score keys ----------------
// grid: (ceil(spatial/1024), 810, 2) ; block 256 ; 4 positions per thread
// Straight-line, wave-uniform async sequence: every lane issues b128 copies with a
// clamped address, wait, read own LDS slots; only the final store is predicated.
__global__ void k_score_cls(const float* __restrict__ cls, uint32_t* __restrict__ keys,
                            int spatial, int pOff) {
    __shared__ __align__(16) float stage[6 * 1024];   // 6 planes x 1024 floats = 24 KB
    const int tid  = threadIdx.x;
    const int cidx = blockIdx.y;
    const int b    = blockIdx.z;
    const int pos  = blockIdx.x * 1024 + tid * 4;
    int posc = pos;
    if (posc > spatial - 4) posc = spatial - 4;       // clamp: uniform issue, dup loads OK

    // low 32 bits of a flat LDS pointer == group-segment byte offset (ISA 10.2)
    const uint32_t so = (uint32_t)(uintptr_t)(&stage[0]) + (uint32_t)(tid * 16);
    const size_t chbase = (size_t)(b * CLS_CH + cidx * 3);

#pragma unroll
    for (int k = 0; k < 3; ++k) {
        const float* gm = cls + (chbase + (size_t)k) * (size_t)spatial + posc;          // mean
        const float* gw = cls + (chbase + (size_t)(4860 + k)) * (size_t)spatial + posc; // weight
        async_g2l_b128(so + (uint32_t)(k * 4096),       gm);
        async_g2l_b128(so + (uint32_t)((k + 3) * 4096), gw);
    }
    wait_async_zero();   // per-wave ASYNCcnt: our own copies are complete & visible

    const float4* sp4 = (const float4*)stage;
    float4 M0 = sp4[0 * 256 + tid], M1 = sp4[1 * 256 + tid], M2 = sp4[2 * 256 + tid];
    float4 W0 = sp4[3 * 256 + tid], W1 = sp4[4 * 256 + tid], W2 = sp4[5 * 256 + tid];

    uint4 kv;
    kv.x = gmm_key(M0.x, M1.x, M2.x, W0.x, W1.x, W2.x);
    kv.y = gmm_key(M0.y, M1.y, M2.y, W0.y, W1.y, W2.y);
    kv.z = gmm_key(M0.z, M1.z, M2.z, W0.z, W1.z, W2.z);
    kv.w = gmm_key(M0.w, M1.w, M2.w, W0.w, W1.w, W2.w);

    if (pos < spatial) {   // pos==posc here; all spatial sizes are multiples of 4
        *(uint4*)&keys[(size_t)b * MPB + (size_t)cidx * PTOT + (size_t)(pOff + pos)] = kv;
    }
}

// ---------------- K2/K4: histograms ----------------
__global__ void k_hist1(const uint32_t* __restrict__ keys, uint32_t* __restrict__ hist) {
    __shared__ uint32_t lh[4096];
    const int tid = threadIdx.x, b = blockIdx.y;
    for (int i = tid; i < 4096; i += 256) lh[i] = 0;
    __syncthreads();
    const uint32_t* kb = keys + (size_t)b * MPB;
    const int stride = gridDim.x * 256;
    for (int i = blockIdx.x * 256 + tid; i < MPB; i += stride) {
        __builtin_prefetch(&kb[i + stride], 0, 0);   // global_prefetch_b8
        atomicAdd(&lh[kb[i] >> 20], 1u);
    }
    __syncthreads();
    for (int i = tid; i < 4096; i += 256) {
        uint32_t c = lh[i];
        if (c) atomicAdd(&hist[b * 4096 + i], c);
    }
}

__global__ void k_hist2(const uint32_t* __restrict__ keys, const uint32_t* __restrict__ misc,
                        uint32_t* __restrict__ hist2) {
    __shared__ uint32_t lh[4096];
    const int tid = threadIdx.x, b = blockIdx.y;
    const uint32_t sel = misc[0 + b];
    for (int i = tid; i < 4096; i += 256) lh[i] = 0;
    __syncthreads();
    const uint32_t* kb = keys + (size_t)b * MPB;
    const int stride = gridDim.x * 256;
    for (int i = blockIdx.x * 256 + tid; i < MPB; i += stride) {
        uint32_t k = kb[i];
        if ((k >> 20) == sel) atomicAdd(&lh[(k >> 8) & 0xFFFu], 1u);
    }
    __syncthreads();
    for (int i = tid; i < 4096; i += 256) {
        uint32_t c = lh[i];
        if (c) atomicAdd(&hist2[b * 4096 + i], c);
    }
}

// ---------------- K3/K5: threshold scans ----------------
// misc: [0..1]=selT  [2..3]=cntAbove  [4..5]=T24  [6..7]=cand count
__global__ void k_scan1(const uint32_t* __restrict__ hist, uint32_t* __restrict__ misc) {
    if (threadIdx.x != 0) return;
    const int b = blockIdx.x;
    uint32_t cum = 0; int sel = 0;
    for (int bin = 4095; bin >= 0; --bin) {
        uint32_t c = hist[b * 4096 + bin];
        if (cum + c >= KDET) { sel = bin; break; }
        cum += c;
    }
    misc[0 + b] = (uint32_t)sel;
    misc[2 + b] = cum;
}

__global__ void k_scan2(const uint32_t* __restrict__ hist2, uint32_t* __restrict__ misc) {
    if (threadIdx.x != 0) return;
    const int b = blockIdx.x;
    uint32_t cum = misc[2 + b];
    uint32_t sel = misc[0 + b];
    int s2 = 0;
    for (int bin = 4095; bin >= 0; --bin) {
        uint32_t c = hist2[b * 4096 + bin];
        if (cum + c >= KDET) { s2 = bin; break; }
        cum += c;
    }
    misc[4 + b] = (sel << 12) | (uint32_t)s2;   // 24-bit threshold prefix
}

// ---------------- K6a: collect candidates >= threshold prefix ----------------
__global__ void k_collect(const uint32_t* __restrict__ keys, uint32_t* __restrict__ misc,
                          uint32_t* __restrict__ candK, uint32_t* __restrict__ candU) {
    const int tid = threadIdx.x, b = blockIdx.y;
    const uint32_t T24 = misc[4 + b];
    const uint32_t* kb = keys + (size_t)b * MPB;
    const int stride = gridDim.x * 256;
    for (int i = blockIdx.x * 256 + tid; i < MPB; i += stride) {
        uint32_t k = kb[i];
        if ((k >> 8) >= T24) {
            uint32_t p = atomicAdd(&misc[6 + b], 1u);
            if (p < CAP) {
                candK[(size_t)b * CAP + p] = k;
                candU[(size_t)b * CAP + p] = (uint32_t)i;
            }
        }
    }
}

// ---------------- K6b: exact rank-by-count (sorted top-K) ----------------
// grid: (64, NBATCH) x 256  -> 16384 threads cover CAP
__global__ void k_rank(const uint32_t* __restrict__ candK, const uint32_t* __restrict__ candU,
                       const uint32_t* __restrict__ misc, uint32_t* __restrict__ outU) {
    __shared__ uint32_t sk[1024], su[1024];
    const int tid = threadIdx.x, b = blockIdx.y;
    uint32_t C = misc[6 + b]; if (C > CAP) C = CAP;
    const int i = blockIdx.x * 256 + tid;
    const bool act = (uint32_t)i < C;
    uint32_t ki = 0, ui = 0;
    if (act) { ki = candK[(size_t)b * CAP + i]; ui = candU[(size_t)b * CAP + i]; }
    uint32_t r = 0;
    for (uint32_t j0 = 0; j0 < C; j0 += 1024) {
        uint32_t cnt = C - j0; if (cnt > 1024) cnt = 1024;
        __syncthreads();
        for (uint32_t t = (uint32_t)tid; t < cnt; t += 256) {
            sk[t] = candK[(size_t)b * CAP + j0 + t];
            su[t] = candU[(size_t)b * CAP + j0 + t];
        }
        __syncthreads();
        if (act) {
            for (uint32_t j = 0; j < cnt; ++j) {
                uint32_t kj = sk[j];
                r += (kj > ki) || (kj == ki && su[j] < ui);   // stable: lower index first
            }
        }
    }
    if (act && r < KDET) outU[(size_t)b * KDET + r] = ui;
}

// ---------------- K7: gather + recompute ua/ue/box at selected anchors ----------------
__global__ void k_gather(LvlPtrs P, const uint32_t* __restrict__ outU, float* __restrict__ out) {
    const int j = blockIdx.x * 256 + threadIdx.x;
    if (j >= BK) return;
    const int b = j / KDET, r = j % KDET;
    const uint32_t u = outU[(size_t)b * KDET + r];
    const int cidx = (int)(u / PTOT);
    const int p    = (int)(u % PTOT);
    const int a = cidx / NCLS, c = cidx % NCLS;

    const int pOff[5] = {0, 4096, 5120, 5376, 5440};
    const int aOff[5] = {0, 36864, 46080, 48384, 48960};
    const int spat[5] = {4096, 1024, 256, 64, 16};
    int l = 4;
    if      (p < 4096) l = 0;
    else if (p < 5120) l = 1;
    else if (p < 5376) l = 2;
    else if (p < 5440) l = 3;
    const int pos = p - pOff[l];
    const int sp  = spat[l];
    const int n   = aOff[l] + pos * NA + a;
    const int o   = b * KDET + r;

    // ---- class GMM stats ----
    {
        const float* cp = P.cls[l];
        size_t base = ((size_t)(b * CLS_CH + cidx * 3)) * (size_t)sp + (size_t)pos;
        float m[3], v[3], w[3];
#pragma unroll
        for (int k = 0; k < 3; ++k) {
            m[k] = cp[base + (size_t)k * sp];
            v[k] = cp[base + (size_t)(2430 + k) * sp];
            w[k] = cp[base + (size_t)(4860 + k) * sp];
        }
        float wmax = fmaxf(w[0], fmaxf(w[1], w[2]));
        float e0 = expf(w[0] - wmax), e1 = expf(w[1] - wmax), e2 = expf(w[2] - wmax);
        float inv = 1.0f / (e0 + e1 + e2);
        float q0 = e0 * inv, q1 = e1 * inv, q2 = e2 * inv;
        float wm = q0 * m[0] + q1 * m[1] + q2 * m[2];
        float ua = q0 * sigm(v[0]) + q1 * sigm(v[1]) + q2 * sigm(v[2]);
        float d0 = m[0] - wm, d1 = m[1] - wm, d2 = m[2] - wm;
        float ue = q0 * d0 * d0 + q1 * d1 * d1 + q2 * d2 * d2;
        out[o]          = wm;   // cls_top
        out[BK + o]     = ua;   // cls_ua_t
        out[2 * BK + o] = ue;   // cls_ue_t
    }

    // ---- box GMM stats ----
    {
        const float* bp = P.box[l];
        float mua = -INFINITY, mue = -INFINITY;
#pragma unroll
        for (int d = 0; d < 4; ++d) {
            const int cb = a * 4 + d;
            size_t b2 = ((size_t)(b * BOX_CH + cb * 3)) * (size_t)sp + (size_t)pos;
            float m[3], v[3], w[3];
#pragma unroll
            for (int k = 0; k < 3; ++k) {
                m[k] = bp[b2 + (size_t)k * sp];
                v[k] = bp[b2 + (size_t)(108 + k) * sp];
                w[k] = bp[b2 + (size_t)(216 + k) * sp];
            }
            float wmax = fmaxf(w[0], fmaxf(w[1], w[2]));
            float e0 = expf(w[0] - wmax), e1 = expf(w[1] - wmax), e2 = expf(w[2] - wmax);
            float inv = 1.0f / (e0 + e1 + e2);
            float q0 = e0 * inv, q1 = e1 * inv, q2 = e2 * inv;
            float wm = q0 * m[0] + q1 * m[1] + q2 * m[2];
            float ua = q0 * sigm(v[0]) + q1 * sigm(v[1]) + q2 * sigm(v[2]);
            float d0 = m[0] - wm, d1 = m[1] - wm, d2 = m[2] - wm;
            float ue = q0 * d0 * d0 + q1 * d1 * d1 + q2 * d2 * d2;
            out[3 * BK + o * 4 + d] = wm;   // box_top
            mua = fmaxf(mua, ua);
            mue = fmaxf(mue, ue);
        }
        out[7 * BK + o] = mua;              // box_ua_t
        out[8 * BK + o] = mue;              // box_ue_t
    }

    out[9 * BK + o]  = (float)n;            // idx
    out[10 * BK + o] = (float)c;            // cls_idx
}

// ---------------- host launch ----------------
extern "C" void kernel_launch(void* const* d_in, const int* in_sizes, int n_in,
                              void* d_out, int out_size, void* d_ws, size_t ws_size,
                              hipStream_t stream) {
    (void)in_sizes; (void)n_in; (void)out_size; (void)ws_size;

    LvlPtrs P;
    for (int i = 0; i < 5; ++i) {
        P.cls[i] = (const float*)d_in[i];
        P.box[i] = (const float*)d_in[5 + i];
    }

    uint8_t* ws = (uint8_t*)d_ws;
    const size_t Z = (size_t)NBATCH * MPB * 4;         // 35,354,880 B of keys
    uint32_t* keys  = (uint32_t*)ws;
    uint32_t* hist1 = (uint32_t*)(ws + Z);             // 2*4096
    uint32_t* hist2 = hist1 + NBATCH * 4096;           // 2*4096
    uint32_t* misc  = hist2 + NBATCH * 4096;           // 64 u32
    uint32_t* outU  = misc + 64;                       // 2*5000
    const size_t zbytes = (size_t)(NBATCH * 4096 * 2 + 64 + NBATCH * KDET) * 4;
    size_t coff = Z + ((zbytes + 255) & ~(size_t)255);
    uint32_t* candK = (uint32_t*)(ws + coff);          // 2*CAP
    uint32_t* candU = candK + NBATCH * CAP;            // 2*CAP

    hipMemsetAsync(ws + Z, 0, zbytes, stream);         // hist/misc/outU must start zeroed

    const int spat[5] = {4096, 1024, 256, 64, 16};
    const int pOff[5] = {0, 4096, 5120, 5376, 5440};
    for (int l = 0; l < 5; ++l) {
        dim3 g((spat[l] + 1023) / 1024, CCLS, NBATCH);
        k_score_cls<<<g, 256, 0, stream>>>(P.cls[l], keys, spat[l], pOff[l]);
    }
    k_hist1<<<dim3(256, NBATCH), 256, 0, stream>>>(keys, hist1);
    k_scan1<<<NBATCH, 32, 0, stream>>>(hist1, misc);
    k_hist2<<<dim3(256, NBATCH), 256, 0, stream>>>(keys, misc, hist2);
    k_scan2<<<NBATCH, 32, 0, stream>>>(hist2, misc);
    k_collect<<<dim3(256, NBATCH), 256, 0, stream>>>(keys, misc, candK, candU);
    k_rank<<<dim3(64, NBATCH), 256, 0, stream>>>(candK, candU, misc, outU);
    k_gather<<<(BK + 255) / 256, 256, 0, stream>>>(P, outU, (float*)d_out);
}